// Frequency_Condition_Attention_10677288698047
// MI455X (gfx1250) — compile-verified
//
#include <hip/hip_runtime.h>

#define BB 8
#define HH 256
#define WW 256
#define HW 65536
#define NHID 64
#define NGROUP 32
#define NDIMX 32
#define NKX 129
#define LL (256*129)   /* 33024 */

typedef float v2f __attribute__((ext_vector_type(2)));
typedef float v8f __attribute__((ext_vector_type(8)));
typedef unsigned int u32x4 __attribute__((ext_vector_type(4)));
typedef int i32x4 __attribute__((ext_vector_type(4)));
typedef int i32x8 __attribute__((ext_vector_type(8)));

// ---------------------------------------------------------------------------
// CDNA5 fp32 WMMA: D(16x16) = A(16x4) * B(4x16) + C   (V_WMMA_F32_16X16X4_F32)
// ---------------------------------------------------------------------------
#if __has_builtin(__builtin_amdgcn_wmma_f32_16x16x4_f32)
__device__ inline v8f wmma_f32(v2f a, v2f b, v8f c) {
  return __builtin_amdgcn_wmma_f32_16x16x4_f32(false, a, false, b, (short)0, c,
                                               false, false);
}
#else
#warning "CDNA5: f32 WMMA builtin unavailable - scalar fallback (wmma=0!)"
__device__ inline v8f wmma_f32(v2f a, v2f b, v8f c) {
#pragma unroll
  for (int r = 0; r < 8; ++r) c[r] += a.x * b.x + a.y * b.y;
  return c;
}
#endif

__device__ inline float fast_sigmoid(float x) {
#if __has_builtin(__builtin_amdgcn_rcpf)
  return __builtin_amdgcn_rcpf(1.f + __expf(-x));
#else
  return 1.f / (1.f + __expf(-x));
#endif
}

__device__ inline int bitrev8(int i) { return (int)(__brev((unsigned)i) >> 24); }

// ---------------------------------------------------------------------------
// Tensor Data Mover: DMA `nrows` rows of 64 f32 (contiguous, stride 64) from
// global to LDS byte-offset `lds_off`.  D# per CDNA5 ISA Sec 8.3-8.4.
// ---------------------------------------------------------------------------
#if __has_builtin(__builtin_amdgcn_tensor_load_to_lds) && \
    __has_builtin(__builtin_amdgcn_s_wait_tensorcnt)
#define HAVE_TDM 1
#else
#warning "CDNA5: tensor_load_to_lds builtin unavailable - cooperative-load fallback in gram kernel"
#endif

__device__ inline void tdm_load_rows64(const float* gsrc, unsigned lds_off,
                                       int nrows) {
#if defined(HAVE_TDM)
  const unsigned long long ga = (unsigned long long)(uintptr_t)gsrc;
  u32x4 g0;
  g0.x = 1u;                                   // count=1 (valid user D#)
  g0.y = lds_off;                              // lds_addr (bytes)
  g0.z = (unsigned)(ga & 0xFFFFFFFFull);       // global_addr[31:0]
  g0.w = (unsigned)((ga >> 32) & 0x1FFFFFFull) // global_addr[56:32]
         | (2u << 30);                         // type=2 ("image")
  i32x8 g1;
  g1[0] = (2 << 16);        // wg_mask=0, data_size=2 (4 bytes)
  g1[1] = (64 << 16);       // tensor_dim0 = 64 (lo16)
  g1[2] = (nrows << 16);    // tensor_dim0 hi=0 | tensor_dim1 lo16 = nrows
  g1[3] = (64 << 16);       // tensor_dim1 hi=0 | tile_dim0 = 64
  g1[4] = nrows;            // tile_dim1 = nrows, tile_dim2 = 0
  g1[5] = 64;               // tensor_dim0_stride lo32 = 64
  g1[6] = 0;
  g1[7] = 0;
  i32x4 z4 = {};
#if __clang_major__ >= 23
  i32x8 z8 = {};
  __builtin_amdgcn_tensor_load_to_lds(g0, g1, z4, z4, z8, 0);
#else
  __builtin_amdgcn_tensor_load_to_lds(g0, g1, z4, z4, 0);
#endif
#else
  (void)gsrc; (void)lds_off; (void)nrows;
#endif
}

// ===========================================================================
// K1: 3x3 conv (8->128, pad=1, no bias) FUSED with GroupNorm partial stats.
// Each block covers 8 out-channels x 256 pixels = 2 groups of 4 channels.
// ===========================================================================
__global__ void conv3x3_stats_kernel(const float* __restrict__ cin,
                                     const float* __restrict__ wconv,
                                     float* __restrict__ q1,
                                     float* __restrict__ gstat) {
  __shared__ float wsm[8 * 72];
  __shared__ float rA[256], rB[256];
  const int tid = threadIdx.x;
  const int ocb = blockIdx.y * 8;
  const int b = blockIdx.z;
  for (int i = tid; i < 8 * 72; i += 256) wsm[i] = wconv[ocb * 72 + i];
  __syncthreads();
  const int p = blockIdx.x * 256 + tid;
  const int y = p >> 8, x = p & 255;
  float vin[72];
#pragma unroll
  for (int ic = 0; ic < 8; ++ic) {
    const float* base = cin + ((size_t)(b * 8 + ic)) * HW;
#pragma unroll
    for (int dy = -1; dy <= 1; ++dy) {
      const int yy = y + dy;
#pragma unroll
      for (int dx = -1; dx <= 1; ++dx) {
        const int xx = x + dx;
        float v = 0.f;
        if ((unsigned)yy < 256u && (unsigned)xx < 256u) v = base[yy * 256 + xx];
        vin[ic * 9 + (dy + 1) * 3 + (dx + 1)] = v;
      }
    }
  }
  float s[2] = {0.f, 0.f}, s2[2] = {0.f, 0.f};
#pragma unroll
  for (int o = 0; o < 8; ++o) {
    float acc = 0.f;
#pragma unroll
    for (int t = 0; t < 72; ++t) acc = fmaf(vin[t], wsm[o * 72 + t], acc);
    q1[((size_t)b * 128 + ocb + o) * HW + p] = acc;
    s[o >> 2] += acc;
    s2[o >> 2] += acc * acc;
  }
  // Block-level reduction of the two 4-channel group partial sums.
  for (int gl = 0; gl < 2; ++gl) {
    __syncthreads();
    rA[tid] = s[gl];
    rB[tid] = s2[gl];
    __syncthreads();
    for (int off = 128; off > 0; off >>= 1) {
      if (tid < off) {
        rA[tid] += rA[tid + off];
        rB[tid] += rB[tid + off];
      }
      __syncthreads();
    }
    if (tid == 0) {
      const int g = (ocb >> 2) + gl;  // group index 0..31
      atomicAdd(&gstat[(b * NGROUP + g) * 2 + 0], rA[0]);
      atomicAdd(&gstat[(b * NGROUP + g) * 2 + 1], rB[0]);
    }
  }
}

// ===========================================================================
// K2: finalize GroupNorm -> per-(b,c) scale/shift.  One tiny block.
// ===========================================================================
__global__ void gn_finalize_kernel(const float* __restrict__ gstat,
                                   const float* __restrict__ gamma,
                                   const float* __restrict__ beta,
                                   float* __restrict__ ss) {
  const int t = threadIdx.x;  // 0..255 = b*32+g
  const int b = t >> 5, g = t & 31;
  const float n = (float)(4 * HW);
  const float mean = gstat[t * 2 + 0] / n;
  const float var = gstat[t * 2 + 1] / n - mean * mean;
  const float rstd = rsqrtf(var + 1e-5f);
#pragma unroll
  for (int j = 0; j < 4; ++j) {
    const int c = g * 4 + j;
    const float sc = gamma[c] * rstd;
    ss[(b * 128 + c) * 2 + 0] = sc;
    ss[(b * 128 + c) * 2 + 1] = beta[c] - mean * sc;
  }
}

// ===========================================================================
// K3: fused GN-apply + SiLU + 1x1 conv (128->64) via f32 WMMA.
// ===========================================================================
__global__ void gnsilu_conv1x1_wmma_kernel(const float* __restrict__ q1,
                                           const float* __restrict__ w2,
                                           const float* __restrict__ b2,
                                           const float* __restrict__ ss,
                                           float* __restrict__ qout) {
  __shared__ float wlds[64 * 128];
  __shared__ float sc[128], sh[128];
  const int tid = threadIdx.x;
  const int b = blockIdx.y;
  for (int i = tid; i < 64 * 128; i += 256) wlds[i] = w2[i];
  for (int i = tid; i < 128; i += 256) {
    sc[i] = ss[(b * 128 + i) * 2 + 0];
    sh[i] = ss[(b * 128 + i) * 2 + 1];
  }
  __syncthreads();
  const int lane = tid & 31, wave = tid >> 5;
  const int n = blockIdx.x * 128 + wave * 16 + (lane & 15);
  const int khalf = (lane >> 4) << 1;
  const float* qb = q1 + (size_t)b * 128 * HW;
  v8f acc[4] = {};
  for (int k0 = 0; k0 < 128; k0 += 4) {
    const int kk = k0 + khalf;
    if (k0 + 4 < 128)
      __builtin_prefetch(&qb[(size_t)(kk + 4) * HW + n], 0, 1);
    const float t0 = qb[(size_t)kk * HW + n] * sc[kk] + sh[kk];
    const float t1 = qb[(size_t)(kk + 1) * HW + n] * sc[kk + 1] + sh[kk + 1];
    v2f bt;
    bt.x = t0 * fast_sigmoid(t0);
    bt.y = t1 * fast_sigmoid(t1);
#pragma unroll
    for (int mt = 0; mt < 4; ++mt) {
      const int m = mt * 16 + (lane & 15);
      v2f at;
      at.x = wlds[m * 128 + kk];
      at.y = wlds[m * 128 + kk + 1];
      acc[mt] = wmma_f32(at, bt, acc[mt]);
    }
  }
#pragma unroll
  for (int mt = 0; mt < 4; ++mt) {
    const int mbase = mt * 16 + (lane >> 4) * 8;
#pragma unroll
    for (int r = 0; r < 8; ++r) {
      const int m = mbase + r;
      qout[((size_t)b * NHID + m) * HW + n] = acc[mt][r] + b2[m];
    }
  }
}

// ===========================================================================
// K4: k = kw@x + kb ; v = vw@x + vb  (1x1 convs, 32->64) via f32 WMMA.
// ===========================================================================
__global__ void kv_conv1x1_wmma_kernel(const float* __restrict__ x,
                                       const float* __restrict__ kw,
                                       const float* __restrict__ kb,
                                       const float* __restrict__ vw,
                                       const float* __restrict__ vb,
                                       float* __restrict__ kout,
                                       float* __restrict__ vout) {
  __shared__ float klds[64 * 32], vlds[64 * 32];
  const int tid = threadIdx.x, b = blockIdx.y;
  for (int i = tid; i < 64 * 32; i += 256) {
    klds[i] = kw[i];
    vlds[i] = vw[i];
  }
  __syncthreads();
  const int lane = tid & 31, wave = tid >> 5;
  const int n = blockIdx.x * 128 + wave * 16 + (lane & 15);
  const int khalf = (lane >> 4) << 1;
  const float* xb = x + (size_t)b * NDIMX * HW;
  v8f ak[4] = {}, av[4] = {};
  for (int k0 = 0; k0 < 32; k0 += 4) {
    const int kk = k0 + khalf;
    v2f bt;
    bt.x = xb[(size_t)kk * HW + n];
    bt.y = xb[(size_t)(kk + 1) * HW + n];
#pragma unroll
    for (int mt = 0; mt < 4; ++mt) {
      const int m = mt * 16 + (lane & 15);
      v2f a1, a2;
      a1.x = klds[m * 32 + kk];
      a1.y = klds[m * 32 + kk + 1];
      ak[mt] = wmma_f32(a1, bt, ak[mt]);
      a2.x = vlds[m * 32 + kk];
      a2.y = vlds[m * 32 + kk + 1];
      av[mt] = wmma_f32(a2, bt, av[mt]);
    }
  }
#pragma unroll
  for (int mt = 0; mt < 4; ++mt) {
    const int mbase = mt * 16 + (lane >> 4) * 8;
#pragma unroll
    for (int r = 0; r < 8; ++r) {
      const int m = mbase + r;
      kout[((size_t)b * NHID + m) * HW + n] = ak[mt][r] + kb[m];
      vout[((size_t)b * NHID + m) * HW + n] = av[mt][r] + vb[m];
    }
  }
}

// ===========================================================================
// FFT butterfly body: 8 radix-2 stages, fully unrolled, shift/mask indexing.
// ===========================================================================
#define FFT_STAGES(reA, imA, lt)                                         \
  _Pragma("unroll") for (int s = 1; s <= 8; ++s) {                       \
    const int hl = 1 << (s - 1);                                         \
    const int pos = (lt) & (hl - 1);                                     \
    const int i0 = (((lt) >> (s - 1)) << s) + pos;                       \
    const int i1 = i0 + hl;                                              \
    const float ang = -6.283185307179586f * (float)pos / (float)(1 << s);\
    float sn, cs;                                                        \
    __sincosf(ang, &sn, &cs);                                            \
    const float ur = (reA)[i0], ui = (imA)[i0];                          \
    const float vr = (reA)[i1], vi = (imA)[i1];                          \
    const float wr = vr * cs - vi * sn;                                  \
    const float wi = vr * sn + vi * cs;                                  \
    (reA)[i0] = ur + wr;                                                 \
    (imA)[i0] = ui + wi;                                                 \
    (reA)[i1] = ur - wr;                                                 \
    (imA)[i1] = ui - wi;                                                 \
    __syncthreads();                                                     \
  }

// K5: 256-pt real FFT along W per row; store transposed [bc][kx][y].
__global__ void fft_row_kernel(const float* __restrict__ src,
                               float* __restrict__ outRe,
                               float* __restrict__ outIm) {
  __shared__ float re[2][256], im[2][256];
  const int tid = threadIdx.x;
  const int hsel = tid >> 7;
  const int lt = tid & 127;
  const long row = (long)blockIdx.x * 2 + hsel;  // (b*64+ch)*256 + y
  const float* rp = src + row * 256;
  re[hsel][bitrev8(lt)] = rp[lt];
  im[hsel][bitrev8(lt)] = 0.f;
  re[hsel][bitrev8(lt + 128)] = rp[lt + 128];
  im[hsel][bitrev8(lt + 128)] = 0.f;
  __syncthreads();
  FFT_STAGES(re[hsel], im[hsel], lt)
  const long bc = row >> 8;
  const int y = (int)(row & 255);
  outRe[(bc * NKX + lt) * 256 + y] = re[hsel][lt];
  outIm[(bc * NKX + lt) * 256 + y] = im[hsel][lt];
  if (lt == 0) {
    outRe[(bc * NKX + 128) * 256 + y] = re[hsel][128];
    outIm[(bc * NKX + 128) * 256 + y] = im[hsel][128];
  }
}

// K6: 256-pt complex FFT along H per (bc,kx) column; scale 1/256 (ortho);
// scatter to spec[b][l][ch]; FUSED per-(b,ch) l-sum via atomicAdd.
__global__ void fft_col_kernel(const float* __restrict__ inRe,
                               const float* __restrict__ inIm,
                               float* __restrict__ specR,
                               float* __restrict__ specI,
                               float* __restrict__ sumR,
                               float* __restrict__ sumI) {
  __shared__ float re[2][256], im[2][256];
  __shared__ float sRred[2][128], sIred[2][128];
  const int tid = threadIdx.x;
  const int hsel = tid >> 7;
  const int lt = tid & 127;
  const long col = (long)blockIdx.x * 2 + hsel;  // (b*64+ch)*129 + kx
  const float* pr = inRe + col * 256;
  const float* pi = inIm + col * 256;
  re[hsel][bitrev8(lt)] = pr[lt];
  im[hsel][bitrev8(lt)] = pi[lt];
  re[hsel][bitrev8(lt + 128)] = pr[lt + 128];
  im[hsel][bitrev8(lt + 128)] = pi[lt + 128];
  __syncthreads();
  FFT_STAGES(re[hsel], im[hsel], lt)
  const long bc = col / NKX;
  const int kx = (int)(col % NKX);
  const long b = bc >> 6;
  const int ch = (int)(bc & 63);
  const float scale = 1.f / 256.f;
  const size_t base = (size_t)b * LL * 64;
  const float r0 = re[hsel][lt] * scale, i0 = im[hsel][lt] * scale;
  const float r1 = re[hsel][lt + 128] * scale, i1 = im[hsel][lt + 128] * scale;
  specR[base + ((size_t)lt * NKX + kx) * 64 + ch] = r0;
  specI[base + ((size_t)lt * NKX + kx) * 64 + ch] = i0;
  specR[base + ((size_t)(lt + 128) * NKX + kx) * 64 + ch] = r1;
  specI[base + ((size_t)(lt + 128) * NKX + kx) * 64 + ch] = i1;
  // column sum over ky -> contributes to sum over l for this (b,ch)
  sRred[hsel][lt] = r0 + r1;
  sIred[hsel][lt] = i0 + i1;
  __syncthreads();
  for (int off = 64; off > 0; off >>= 1) {
    if (lt < off) {
      sRred[hsel][lt] += sRred[hsel][lt + off];
      sIred[hsel][lt] += sIred[hsel][lt + off];
    }
    __syncthreads();
  }
  if (lt == 0) {
    atomicAdd(&sumR[bc], sRred[hsel][0]);
    atomicAdd(&sumI[bc], sIred[hsel][0]);
  }
}

// ===========================================================================
// K8: Gram matrices via WMMA, TDM double-buffered LDS staging, A shared
// between two Grams.  qsel=0: A=QR -> G0(rr, B=KR) and G2(ri, B=KI).
//                     qsel=1: A=QI -> G3(ir, B=KR) and G1(ii, B=KI).
// ===========================================================================
#define GRAM_SPLIT 129
#define GRAM_ROWS 16
#define GRAM_CHUNKS 16  /* 16*16 = 256 rows per block */

__global__ void gram_wmma_kernel(const float* __restrict__ QR,
                                 const float* __restrict__ QI,
                                 const float* __restrict__ KR,
                                 const float* __restrict__ KI,
                                 float* __restrict__ G) {
  // [2 buf][A | B0 | B1] each 16x64 f32 -> 6144 floats (24 KB), at LDS 0.
  __shared__ float lds_all[2 * 3 * 1024];
  const int b = blockIdx.z;
  const int qsel = blockIdx.y;
  const int zs = blockIdx.x;
  const float* Asp = (qsel == 0 ? QR : QI) + (size_t)b * LL * 64;
  const float* Bsp0 = KR + (size_t)b * LL * 64;
  const float* Bsp1 = KI + (size_t)b * LL * 64;
  const int gid0 = (qsel == 0) ? 0 : 3;
  const int gid1 = (qsel == 0) ? 2 : 1;
  const long kbeg = (long)zs * (LL / GRAM_SPLIT);  // *256
  const int lane = threadIdx.x & 31, wave = threadIdx.x >> 5;
  const int khalf = (lane >> 4) << 1;
  v8f acc[2][2] = {};  // [tile][gram]
  int mt[2], nt[2];
#pragma unroll
  for (int t = 0; t < 2; ++t) {
    const int tt = wave * 2 + t;
    mt[t] = tt >> 2;
    nt[t] = tt & 3;
  }
#if defined(HAVE_TDM)
  if (wave == 0) {
    tdm_load_rows64(Asp + kbeg * 64, 0u, GRAM_ROWS);
    tdm_load_rows64(Bsp0 + kbeg * 64, 4096u, GRAM_ROWS);
    tdm_load_rows64(Bsp1 + kbeg * 64, 8192u, GRAM_ROWS);
    __builtin_amdgcn_s_wait_tensorcnt(0);
  }
  __syncthreads();
#endif
  for (int c = 0; c < GRAM_CHUNKS; ++c) {
    const int cur = c & 1;
#if defined(HAVE_TDM)
    const int nxt = cur ^ 1;
    if (wave == 0 && c + 1 < GRAM_CHUNKS) {
      const long rb = kbeg + (long)(c + 1) * GRAM_ROWS;
      const unsigned bb = (unsigned)(nxt * 12288);
      tdm_load_rows64(Asp + rb * 64, bb, GRAM_ROWS);
      tdm_load_rows64(Bsp0 + rb * 64, bb + 4096u, GRAM_ROWS);
      tdm_load_rows64(Bsp1 + rb * 64, bb + 8192u, GRAM_ROWS);
    }
#else
    const long rb0 = kbeg + (long)c * GRAM_ROWS;
    for (int i = threadIdx.x; i < GRAM_ROWS * 64; i += 256) {
      lds_all[cur * 3072 + i] = Asp[rb0 * 64 + i];
      lds_all[cur * 3072 + 1024 + i] = Bsp0[rb0 * 64 + i];
      lds_all[cur * 3072 + 2048 + i] = Bsp1[rb0 * 64 + i];
    }
    __syncthreads();
#endif
    const float* la = lds_all + cur * 3072;
    const float* lb0 = la + 1024;
    const float* lb1 = la + 2048;
#pragma unroll
    for (int k0 = 0; k0 < GRAM_ROWS; k0 += 4) {
      const int kl = k0 + khalf;
#pragma unroll
      for (int t = 0; t < 2; ++t) {
        const int m = mt[t] * 16 + (lane & 15);
        const int n = nt[t] * 16 + (lane & 15);
        v2f at, bt0, bt1;
        at.x = la[kl * 64 + m];
        at.y = la[(kl + 1) * 64 + m];
        bt0.x = lb0[kl * 64 + n];
        bt0.y = lb0[(kl + 1) * 64 + n];
        bt1.x = lb1[kl * 64 + n];
        bt1.y = lb1[(kl + 1) * 64 + n];
        acc[t][0] = wmma_f32(at, bt0, acc[t][0]);
        acc[t][1] = wmma_f32(at, bt1, acc[t][1]);
      }
    }
#if defined(HAVE_TDM)
    if (wave == 0 && c + 1 < GRAM_CHUNKS) __builtin_amdgcn_s_wait_tensorcnt(0);
#endif
    __syncthreads();
  }
  float* Gp0 = G + ((size_t)b * 4 + gid0) * 4096;
  float* Gp1 = G + ((size_t)b * 4 + gid1) * 4096;
#pragma unroll
  for (int t = 0; t < 2; ++t) {
    const int n = nt[t] * 16 + (lane & 15);
    const int mbase = mt[t] * 16 + (lane >> 4) * 8;
#pragma unroll
    for (int r = 0; r < 8; ++r) {
      atomicAdd(&Gp0[(mbase + r) * 64 + n], acc[t][0][r]);
      atomicAdd(&Gp1[(mbase + r) * 64 + n], acc[t][1][r]);
    }
  }
}

// ===========================================================================
// K9: assemble frequency-domain A (64x64 complex) per batch.
// ===========================================================================
__global__ void assemble_A_kernel(
    const float* __restrict__ G, const float* __restrict__ sums,
    const float* __restrict__ Wqr, const float* __restrict__ bqr,
    const float* __restrict__ Wqi, const float* __restrict__ bqi,
    const float* __restrict__ Wkr, const float* __restrict__ bkr,
    const float* __restrict__ Wki, const float* __restrict__ bki,
    float* __restrict__ AfRe, float* __restrict__ AfIm) {
  const int b = blockIdx.x;
  const int tid = threadIdx.x;
  __shared__ float u1[64], u2[64], w1[64], w2[64];
  __shared__ float tmp[64 * 64];
  __shared__ float are[64 * 64], aim[64 * 64];
  const int SB = BB * 64;
  const float* sqr = sums + 0 * SB + b * 64;
  const float* sqi = sums + 1 * SB + b * 64;
  const float* skr = sums + 2 * SB + b * 64;
  const float* ski = sums + 3 * SB + b * 64;
  {
    const int which = tid >> 6;
    const int c = tid & 63;
    const float* Wm;
    const float* vec;
    float* dst;
    if (which == 0) { Wm = Wqr; vec = sqr; dst = u1; }
    else if (which == 1) { Wm = Wqi; vec = sqi; dst = u2; }
    else if (which == 2) { Wm = Wkr; vec = skr; dst = w1; }
    else { Wm = Wki; vec = ski; dst = w2; }
    float s = 0.f;
    for (int e = 0; e < 64; ++e) s += Wm[c * 64 + e] * vec[e];
    dst[c] = s;
  }
  __syncthreads();
  const float Lf = (float)LL;
  for (int i = tid; i < 4096; i += 256) {
    const int c = i >> 6, d = i & 63;
    are[i] = bqr[c] * w1[d] + bkr[d] * u1[c] + Lf * bqr[c] * bkr[d] -
             (bqi[c] * w2[d] + bki[d] * u2[c] + Lf * bqi[c] * bki[d]);
    aim[i] = bqr[c] * w2[d] + bki[d] * u1[c] + Lf * bqr[c] * bki[d] +
             bqi[c] * w1[d] + bkr[d] * u2[c] + Lf * bqi[c] * bkr[d];
  }
  __syncthreads();
  for (int pass = 0; pass < 4; ++pass) {
    const float* Gp = G + ((size_t)b * 4 + pass) * 4096;
    const float* Wq = (pass == 0 || pass == 2) ? Wqr : Wqi;
    const float* Wk = (pass == 0 || pass == 3) ? Wkr : Wki;
    for (int i = tid; i < 4096; i += 256) {
      const int c = i >> 6, f = i & 63;
      float s = 0.f;
      for (int e = 0; e < 64; ++e) s += Wq[c * 64 + e] * Gp[e * 64 + f];
      tmp[i] = s;
    }
    __syncthreads();
    for (int i = tid; i < 4096; i += 256) {
      const int c = i >> 6, d = i & 63;
      float s = 0.f;
      for (int f = 0; f < 64; ++f) s += tmp[c * 64 + f] * Wk[d * 64 + f];
      if (pass == 0) are[i] += s;
      else if (pass == 1) are[i] -= s;
      else aim[i] += s;
    }
    __syncthreads();
  }
  for (int i = tid; i < 4096; i += 256) {
    AfRe[(size_t)b * 4096 + i] = are[i];
    AfIm[(size_t)b * 4096 + i] = aim[i];
  }
}

// ===========================================================================
// K10: irfft2(A, s=(64,64), ortho) (truncates last axis 64->33) + softmax.
// ===========================================================================
__global__ void irfft_softmax_kernel(const float* __restrict__ AfRe,
                                     const float* __restrict__ AfIm,
                                     float* __restrict__ Aout) {
  const int b = blockIdx.x;
  const int tid = threadIdx.x;
  __shared__ float mr[64 * 33], mi[64 * 33];
  __shared__ float S[64 * 64];
  const float* ar = AfRe + (size_t)b * 4096;
  const float* ai = AfIm + (size_t)b * 4096;
  const float TWO_PI = 6.283185307179586f;
  for (int i = tid; i < 64 * 33; i += 256) {
    const int m = i / 33, k = i % 33;
    float sr = 0.f, si = 0.f;
    for (int j = 0; j < 64; ++j) {
      const float ang = TWO_PI * (float)((j * m) & 63) / 64.f;
      float sn, cs;
      __sincosf(ang, &sn, &cs);
      const float xr = ar[j * 64 + k], xi = ai[j * 64 + k];
      sr += xr * cs - xi * sn;
      si += xr * sn + xi * cs;
    }
    mr[i] = sr * 0.125f;
    mi[i] = si * 0.125f;
  }
  __syncthreads();
  for (int i = tid; i < 4096; i += 256) {
    const int m = i >> 6, n = i & 63;
    float s = mr[m * 33 + 0];
    s += ((n & 1) ? -1.f : 1.f) * mr[m * 33 + 32];
    for (int k = 1; k < 32; ++k) {
      const float ang = TWO_PI * (float)((k * n) & 63) / 64.f;
      float sn, cs;
      __sincosf(ang, &sn, &cs);
      s += 2.f * (mr[m * 33 + k] * cs - mi[m * 33 + k] * sn);
    }
    S[i] = s * 0.125f;
  }
  __syncthreads();
  if (tid < 64) {
    float mx = -1e30f;
    for (int n = 0; n < 64; ++n) mx = fmaxf(mx, S[tid * 64 + n]);
    float se = 0.f;
    for (int n = 0; n < 64; ++n) se += __expf(S[tid * 64 + n] - mx);
    const float inv = 1.f / se;
    for (int n = 0; n < 64; ++n)
      Aout[(size_t)b * 4096 + tid * 64 + n] = __expf(S[tid * 64 + n] - mx) * inv;
  }
}

// ===========================================================================
// K11: out[b][c][p] = sum_d A[b][c][d] * v[b][d][p]  (WMMA, M=64,K=64,N=HW)
// ===========================================================================
__global__ void out_gemm_wmma_kernel(const float* __restrict__ Aout,
                                     const float* __restrict__ vbuf,
                                     float* __restrict__ out) {
  __shared__ float alds[64 * 64];
  const int tid = threadIdx.x, b = blockIdx.y;
  for (int i = tid; i < 4096; i += 256) alds[i] = Aout[(size_t)b * 4096 + i];
  __syncthreads();
  const int lane = tid & 31, wave = tid >> 5;
  const int n = blockIdx.x * 128 + wave * 16 + (lane & 15);
  const int khalf = (lane >> 4) << 1;
  const float* vp = vbuf + (size_t)b * NHID * HW;
  v8f acc[4] = {};
  for (int k0 = 0; k0 < 64; k0 += 4) {
    const int kk = k0 + khalf;
    if (k0 + 4 < 64)
      __builtin_prefetch(&vp[(size_t)(kk + 4) * HW + n], 0, 1);
    v2f bt;
    bt.x = vp[(size_t)kk * HW + n];
    bt.y = vp[(size_t)(kk + 1) * HW + n];
#pragma unroll
    for (int mt = 0; mt < 4; ++mt) {
      const int m = mt * 16 + (lane & 15);
      v2f at;
      at.x = alds[m * 64 + kk];
      at.y = alds[m * 64 + kk + 1];
      acc[mt] = wmma_f32(at, bt, acc[mt]);
    }
  }
#pragma unroll
  for (int mt = 0; mt < 4; ++mt) {
    const int mbase = mt * 16 + (lane >> 4) * 8;
#pragma unroll
    for (int r = 0; r < 8; ++r)
      out[((size_t)b * NHID + mbase + r) * HW + n] = acc[mt][r];
  }
}

// ===========================================================================
extern "C" void kernel_launch(void* const* d_in, const int* in_sizes, int n_in,
                              void* d_out, int out_size, void* d_ws,
                              size_t ws_size, hipStream_t stream) {
  (void)in_sizes; (void)n_in; (void)out_size; (void)ws_size;
  const float* x = (const float*)d_in[0];
  const float* cin = (const float*)d_in[1];
  const float* qc1w = (const float*)d_in[2];
  const float* gam = (const float*)d_in[3];
  const float* bet = (const float*)d_in[4];
  const float* qc2w = (const float*)d_in[5];
  const float* qc2b = (const float*)d_in[6];
  const float* kw = (const float*)d_in[7];
  const float* kb = (const float*)d_in[8];
  const float* vw = (const float*)d_in[9];
  const float* vb = (const float*)d_in[10];
  const float* Wqr = (const float*)d_in[11];
  const float* bqr = (const float*)d_in[12];
  const float* Wqi = (const float*)d_in[13];
  const float* bqi = (const float*)d_in[14];
  const float* Wkr = (const float*)d_in[15];
  const float* bkr = (const float*)d_in[16];
  const float* Wki = (const float*)d_in[17];
  const float* bki = (const float*)d_in[18];
  float* out = (float*)d_out;

  float* w = (float*)d_ws;
  size_t off = 0;
  float* q1 = w + off;    off += (size_t)BB * 128 * HW;
  float* qbuf = w + off;  off += (size_t)BB * NHID * HW;
  float* kbuf = w + off;  off += (size_t)BB * NHID * HW;
  float* vbuf = w + off;  off += (size_t)BB * NHID * HW;
  float* frowR = w + off; off += (size_t)BB * 64 * NKX * 256;
  float* frowI = w + off; off += (size_t)BB * 64 * NKX * 256;
  float* QRs = w + off;   off += (size_t)BB * LL * 64;
  float* QIs = w + off;   off += (size_t)BB * LL * 64;
  float* KRs = w + off;   off += (size_t)BB * LL * 64;
  float* KIs = w + off;   off += (size_t)BB * LL * 64;
  float* ss = w + off;    off += (size_t)BB * 128 * 2;
  float* sums = w + off;  off += (size_t)4 * BB * 64;
  float* G = w + off;     off += (size_t)BB * 4 * 4096;
  float* AfRe = w + off;  off += (size_t)BB * 4096;
  float* AfIm = w + off;  off += (size_t)BB * 4096;
  float* Asm = w + off;   off += (size_t)BB * 4096;
  float* gstat = w + off; off += (size_t)BB * NGROUP * 2;

  hipMemsetAsync(sums, 0, (size_t)4 * BB * 64 * sizeof(float), stream);
  hipMemsetAsync(G, 0, (size_t)BB * 4 * 4096 * sizeof(float), stream);
  hipMemsetAsync(gstat, 0, (size_t)BB * NGROUP * 2 * sizeof(float), stream);

  conv3x3_stats_kernel<<<dim3(HW / 256, 16, BB), 256, 0, stream>>>(cin, qc1w,
                                                                   q1, gstat);
  gn_finalize_kernel<<<dim3(1), 256, 0, stream>>>(gstat, gam, bet, ss);
  gnsilu_conv1x1_wmma_kernel<<<dim3(HW / 128, BB), 256, 0, stream>>>(
      q1, qc2w, qc2b, ss, qbuf);
  kv_conv1x1_wmma_kernel<<<dim3(HW / 128, BB), 256, 0, stream>>>(
      x, kw, kb, vw, vb, kbuf, vbuf);
  fft_row_kernel<<<dim3(BB * 64 * 256 / 2), 256, 0, stream>>>(qbuf, frowR, frowI);
  fft_col_kernel<<<dim3(BB * 64 * NKX / 2), 256, 0, stream>>>(
      frowR, frowI, QRs, QIs, sums + 0 * BB * 64, sums + 1 * BB * 64);
  fft_row_kernel<<<dim3(BB * 64 * 256 / 2), 256, 0, stream>>>(kbuf, frowR, frowI);
  fft_col_kernel<<<dim3(BB * 64 * NKX / 2), 256, 0, stream>>>(
      frowR, frowI, KRs, KIs, sums + 2 * BB * 64, sums + 3 * BB * 64);
  gram_wmma_kernel<<<dim3(GRAM_SPLIT, 2, BB), 256, 0, stream>>>(QRs, QIs, KRs,
                                                                KIs, G);
  assemble_A_kernel<<<dim3(BB), 256, 0, stream>>>(G, sums, Wqr, bqr, Wqi, bqi,
                                                  Wkr, bkr, Wki, bki, AfRe, AfIm);
  irfft_softmax_kernel<<<dim3(BB), 256, 0, stream>>>(AfRe, AfIm, Asm);
  out_gemm_wmma_kernel<<<dim3(HW / 128, BB), 256, 0, stream>>>(Asm, vbuf, out);
}